// SparseHourglass_61856118997460
// MI455X (gfx1250) — compile-verified
//
#include <hip/hip_runtime.h>
#include <stdint.h>

// SparseHourglass (sparsemax with mass 1-LAM) — MI455X / gfx1250, wave32.
// One 512-thread block (16 waves) per row of 16384 f32.
// Path: async global->LDS (b128, ASYNCcnt, TH_LOAD_NT) -> registers ->
// histogram select of the sparsemax threshold (no sort) -> exact tiny solve ->
// non-temporal streaming b128 stores. One HBM touch per element each way.

#define DIM   16384
#define TPB   512
#define NW    (TPB / 32)     // 16 waves
#define VPT   8              // float4 chunks per thread (8*16B*512 = 64KB row)
#define EPT   32             // scalar elements per thread
#define NB    256            // histogram buckets
#define CAP   128            // compacted candidate capacity

typedef float v4f __attribute__((ext_vector_type(4)));

struct SharedBlk {
  float stage[DIM];          // 64 KB row staging (async landing zone)
  float hcnt[NB];            // bucket counts (float: exact for counts < 2^24)
  float hsum[NB];            // bucket sums
  float wsum[NW];
  float wmax[NW];
  float bufv[CAP];           // compacted in-bucket candidates
  int   sel_t;               // winning bucket (-1 = full support fallback)
  float sel_hc;              // count of elements strictly above winning bucket
  float sel_hs;              // sum   of elements strictly above winning bucket
  int   m;                   // compacted count
  float tau;                 // broadcast threshold
};

__device__ __forceinline__ float wave_red_sum(float v) {
#pragma unroll
  for (int off = 16; off > 0; off >>= 1) v += __shfl_xor(v, off, 32);
  return v;
}
__device__ __forceinline__ float wave_red_max(float v) {
#pragma unroll
  for (int off = 16; off > 0; off >>= 1) v = fmaxf(v, __shfl_xor(v, off, 32));
  return v;
}

__global__ __launch_bounds__(TPB) void sparse_hourglass_kernel(
    const float* __restrict__ x, float* __restrict__ out) {
  __shared__ SharedBlk sh;
  const int tid = threadIdx.x;
  const int lane = tid & 31;
  const int wid = tid >> 5;
  const float* __restrict__ xrow = x + (size_t)blockIdx.x * DIM;
  float* __restrict__ orow = out + (size_t)blockIdx.x * DIM;

  // ---------- Phase 1: async copy row -> LDS (gfx1250 ASYNCcnt path) ----------
  // Input is streamed exactly once: TH_LOAD_NT keeps it from occupying L2.
  {
    const unsigned ldsOff = (unsigned)(uintptr_t)(&sh.stage[0]) + (unsigned)(tid * 16);
    const unsigned gOff   = (unsigned)(tid * 16);
    const unsigned long long gbase = (unsigned long long)(uintptr_t)xrow;
    // GVS mode: the immediate offset is added to both the LDS and the global address.
#define ASYNC_COPY_16B(IMM)                                                     \
    asm volatile("global_load_async_to_lds_b128 %0, %1, %2 offset:" IMM         \
                 " th:TH_LOAD_NT"                                               \
                 :: "v"(ldsOff), "v"(gOff), "s"(gbase) : "memory")
    ASYNC_COPY_16B("0");     ASYNC_COPY_16B("8192");
    ASYNC_COPY_16B("16384"); ASYNC_COPY_16B("24576");
    ASYNC_COPY_16B("32768"); ASYNC_COPY_16B("40960");
    ASYNC_COPY_16B("49152"); ASYNC_COPY_16B("57344");
#undef ASYNC_COPY_16B
  }
  // Zero scratch while the async copies are in flight.
  for (int i = tid; i < NB; i += TPB) { sh.hcnt[i] = 0.0f; sh.hsum[i] = 0.0f; }
  if (tid == 0) { sh.sel_t = -1; sh.sel_hc = 0.0f; sh.sel_hs = 0.0f; sh.m = 0; }
  asm volatile("s_wait_asynccnt 0x0" ::: "memory");
  __syncthreads();

  // ---------- Phase 2: LDS -> registers; block sum & max ----------
  float z[EPT];
  {
    const v4f* ls = (const v4f*)sh.stage;
#pragma unroll
    for (int p = 0; p < VPT; ++p) {
      v4f v = ls[tid + p * TPB];
      z[4 * p + 0] = v.x; z[4 * p + 1] = v.y;
      z[4 * p + 2] = v.z; z[4 * p + 3] = v.w;
    }
  }
  float psum = 0.0f, pmax = -__builtin_inff();
#pragma unroll
  for (int i = 0; i < EPT; ++i) { psum += z[i]; pmax = fmaxf(pmax, z[i]); }
  psum = wave_red_sum(psum);
  pmax = wave_red_max(pmax);
  if (lane == 0) { sh.wsum[wid] = psum; sh.wmax[wid] = pmax; }
  __syncthreads();
  float S = 0.0f, xmax = -__builtin_inff();
#pragma unroll
  for (int i = 0; i < NW; ++i) { S += sh.wsum[i]; xmax = fmaxf(xmax, sh.wmax[i]); }

  // ---------- Phase 3: rescale; histogram of z over tau's provable range ----------
  const float R = 0.95f;                                 // 1 - LAM
  const float dq = (float)DIM * 0.1f;                    // dim * Q
  const float alpha = (1.0f + dq) / (fabsf(S) + dq);
#pragma unroll
  for (int i = 0; i < EPT; ++i) z[i] *= alpha;
  const float Sz = alpha * S;
  const float zmax = alpha * xmax;
  const float lo = (Sz - R) * (1.0f / (float)DIM);       // f(lo) >= R  => tau >= lo
  float w = (zmax - lo) * (1.0f / (float)NB);            // f(zmax) = 0 => tau < zmax
  if (!(w > 0.0f)) w = 1e-30f;
  const float invw = 1.0f / w;

#pragma unroll
  for (int i = 0; i < EPT; ++i) {
    float v = z[i];
    if (v >= lo) {                                       // values < lo never in support
      int t = (int)((v - lo) * invw);
      t = t > (NB - 1) ? (NB - 1) : t;
      atomicAdd(&sh.hcnt[t], 1.0f);                      // ds_add_f32
      atomicAdd(&sh.hsum[t], v);
    }
  }
  __syncthreads();

  // ---------- Phase 4: wave-0 suffix scan finds bucket containing tau ----------
  // f(B) = sum_{z>=B}(z-B) = HS(B) - B*HC(B), monotone decreasing; tau solves f=R.
  if (tid < 32) {
    float c[8], s8[8], ctot = 0.0f, stot = 0.0f;
#pragma unroll
    for (int u = 0; u < 8; ++u) {
      c[u] = sh.hcnt[tid * 8 + u]; s8[u] = sh.hsum[tid * 8 + u];
      ctot += c[u]; stot += s8[u];
    }
    float cs = ctot, ss = stot;                          // inclusive suffix across lanes
#pragma unroll
    for (int off = 1; off < 32; off <<= 1) {
      float oc = __shfl_down(cs, off, 32);
      float os = __shfl_down(ss, off, 32);
      if (tid + off < 32) { cs += oc; ss += os; }
    }
    float hc = cs - ctot, hs = ss - stot;                // buckets owned by lanes above
#pragma unroll
    for (int u = 7; u >= 0; --u) {
      const int t = tid * 8 + u;
      const float Bhi = lo + (float)(t + 1) * w;
      const float Blo = lo + (float)t * w;
      const float fhi = hs - Bhi * hc;                    // f at top of bucket t
      const float flo = (hs + s8[u]) - Blo * (hc + c[u]); // f at bottom of bucket t
      if (flo > R && fhi <= R) { sh.sel_t = t; sh.sel_hc = hc; sh.sel_hs = hs; }
      hc += c[u]; hs += s8[u];
    }
  }
  __syncthreads();

  // ---------- Phase 5: compact the few candidates inside the winning bucket ----------
  const int selt = sh.sel_t;
#pragma unroll
  for (int i = 0; i < EPT; ++i) {
    float v = z[i];
    if (selt >= 0 && v >= lo) {
      int t = (int)((v - lo) * invw);
      t = t > (NB - 1) ? (NB - 1) : t;
      if (t == selt) {
        int p = atomicAdd(&sh.m, 1);
        if (p < CAP) sh.bufv[p] = v;
      }
    }
  }
  __syncthreads();

  // ---------- Phase 6: exact tiny solve (reference support recurrence) ----------
  if (tid == 0) {
    if (selt < 0) {
      sh.tau = (Sz - R) * (1.0f / (float)DIM);           // full-support fallback
    } else {
      int M = sh.m; M = M > CAP ? CAP : M;
      for (int i = 1; i < M; ++i) {                      // descending insertion sort
        float key = sh.bufv[i]; int j = i - 1;
        while (j >= 0 && sh.bufv[j] < key) { sh.bufv[j + 1] = sh.bufv[j]; --j; }
        sh.bufv[j + 1] = key;
      }
      float k = sh.sel_hc, cum = sh.sel_hs;              // all elements above bucket pass
      float kz = k, tausum = cum;
      for (int j = 0; j < M; ++j) {                      // (1-LAM + k*z_(k)) > cumsum_k
        float v = sh.bufv[j];
        cum += v; k += 1.0f;
        if (R + k * v > cum) { kz = k; tausum = cum; }
      }
      if (kz < 1.0f) kz = 1.0f;
      sh.tau = (tausum - R) / kz;
    }
  }
  __syncthreads();

  // ---------- Phase 7: prob = clip(z - tau, 0)/(1-LAM); non-temporal b128 stores ----------
  const float tau = sh.tau;
  const float invR = 1.0f / R;
  v4f* os = (v4f*)orow;
#pragma unroll
  for (int p = 0; p < VPT; ++p) {
    v4f o;
    o.x = fmaxf(z[4 * p + 0] - tau, 0.0f) * invR;
    o.y = fmaxf(z[4 * p + 1] - tau, 0.0f) * invR;
    o.z = fmaxf(z[4 * p + 2] - tau, 0.0f) * invR;
    o.w = fmaxf(z[4 * p + 3] - tau, 0.0f) * invR;
    __builtin_nontemporal_store(o, &os[tid + p * TPB]);  // write-once output: bypass L2 retention
  }
}

extern "C" void kernel_launch(void* const* d_in, const int* in_sizes, int n_in,
                              void* d_out, int out_size, void* d_ws, size_t ws_size,
                              hipStream_t stream) {
  (void)n_in; (void)out_size; (void)d_ws; (void)ws_size;
  const float* x = (const float*)d_in[0];
  float* out = (float*)d_out;
  const int rows = in_sizes[0] / DIM;
  sparse_hourglass_kernel<<<rows, TPB, 0, stream>>>(x, out);
}